// AutoCorrelation_40810779246670
// MI455X (gfx1250) — compile-verified
//
#include <hip/hip_runtime.h>
#include <math.h>

// Problem dims (fixed by the reference)
#define BB   4
#define LL   4096
#define DD   1024
#define HH   16
#define DKK  64
#define MM   (BB * LL)     // 16384 rows
#define TOPK 8

typedef __attribute__((ext_vector_type(2))) float v2f;
typedef __attribute__((ext_vector_type(8))) float v8f;

__device__ __forceinline__ v8f wmma_f32x4(v2f a, v2f b, v8f c) {
  // V_WMMA_F32_16X16X4_F32 : D = A(16x4) * B(4x16) + C(16x16), fp32 exact
  return __builtin_amdgcn_wmma_f32_16x16x4_f32(
      /*neg_a=*/false, a, /*neg_b=*/false, b,
      /*c_mod=*/(short)0, c, /*reuse_a=*/false, /*reuse_b=*/false);
}

// CDNA5 async global->LDS copy (no VGPR round-trip, tracked on ASYNCcnt)
__device__ __forceinline__ void async_load_b128(unsigned lds_off, const void* gaddr) {
  asm volatile("global_load_async_to_lds_b128 %0, %1, off"
               :
               : "v"(lds_off), "v"((unsigned long long)(uintptr_t)gaddr)
               : "memory");
}
__device__ __forceinline__ void wait_async0() {
  asm volatile("s_wait_asynccnt 0x0" ::: "memory");
}

// ---------------------------------------------------------------------------
// 1) Reduce projection weights across each head's 64 channels (mean is linear)
//    Wred[j][h] = mean_c W[j, h*64+c] ; bred[h] = mean_c b[h*64+c]
// ---------------------------------------------------------------------------
__global__ void reduce_w_kernel(const float* __restrict__ W,
                                const float* __restrict__ bvec,
                                float* __restrict__ Wred,
                                float* __restrict__ bred) {
  int idx = blockIdx.x * 256 + threadIdx.x;   // 0 .. 16383
  int j = idx >> 4;       // 0..1023
  int h = idx & 15;       // 0..15
  const float inv = 1.0f / (float)DKK;
  float s = 0.f;
#pragma unroll 8
  for (int c = 0; c < DKK; ++c) s += W[j * DD + h * DKK + c];
  Wred[j * HH + h] = s * inv;
  if (j == 0) {
    float sb = 0.f;
#pragma unroll 8
    for (int c = 0; c < DKK; ++c) sb += bvec[h * DKK + c];
    bred[h] = sb * inv;
  }
}

// ---------------------------------------------------------------------------
// 2) Narrow GEMM with WMMA + async double-buffered staging
//    out(b,h,l) = head-mean-projected sequence
//    X: (MM x DD) row-major, Wred: (DD x HH), out stored as (B*H, L)
//    grid 128 blocks x 256 threads (8 waves, each owns one 16-row tile)
// ---------------------------------------------------------------------------
#define KCM 32
#define PXM 36   // 144B row stride: 16B aligned, column reads conflict-free
#define PWM 20   // 80B row stride: 16B aligned, half-waves bank-disjoint

__global__ void gemm_mean_wmma(const float* __restrict__ X,
                               const float* __restrict__ Wred,
                               const float* __restrict__ bred,
                               float* __restrict__ out) {
  __shared__ float sX[2][128][PXM];   // 36.9 KB
  __shared__ float sW[2][KCM][PWM];   //  5.1 KB

  const int tid  = threadIdx.x;
  const int lane = tid & 31;
  const int wave = tid >> 5;               // 0..7 -> 16-row tile each
  const int blockM = blockIdx.x * 128;

  const int ml   = lane & 15;
  const int koff = (lane >> 4) << 1;       // 0 or 2 (K pair per half-wave)

  // staging: X tile 128x32 = 1024 B128 -> 4 per thread
  const int xr = tid >> 3;                 // 0..31 (+32 per rep)
  const int xc = (tid & 7) * 4;
  // W tile 32x16 = 128 B128 -> waves 0..3 only (wave-uniform predicate)
  const bool wact = tid < 128;
  const int wr = tid >> 2;
  const int wc = (tid & 3) * 4;

  v8f acc = {};
  const int nchunks = DD / KCM;            // 32

  // prologue: stage chunk 0 into buffer 0
#pragma unroll
  for (int rr = 0; rr < 4; ++rr) {
    int r = xr + rr * 32;
    async_load_b128((unsigned)(uintptr_t)&sX[0][r][xc],
                    &X[(size_t)(blockM + r) * DD + 0 + xc]);
  }
  if (wact)
    async_load_b128((unsigned)(uintptr_t)&sW[0][wr][wc],
                    &Wred[(size_t)(0 + wr) * HH + wc]);
  wait_async0();
  __syncthreads();

  for (int c = 0; c < nchunks; ++c) {
    const int cur = c & 1;
    if (c + 1 < nchunks) {
      const int nb = cur ^ 1;
      const int k0 = (c + 1) * KCM;
#pragma unroll
      for (int rr = 0; rr < 4; ++rr) {
        int r = xr + rr * 32;
        async_load_b128((unsigned)(uintptr_t)&sX[nb][r][xc],
                        &X[(size_t)(blockM + r) * DD + k0 + xc]);
      }
      if (wact)
        async_load_b128((unsigned)(uintptr_t)&sW[nb][wr][wc],
                        &Wred[(size_t)(k0 + wr) * HH + wc]);
    }
#pragma unroll
    for (int ks = 0; ks < KCM; ks += 4) {
      v2f a, b;
      a.x = sX[cur][wave * 16 + ml][ks + koff];
      a.y = sX[cur][wave * 16 + ml][ks + koff + 1];
      b.x = sW[cur][ks + koff][ml];
      b.y = sW[cur][ks + koff + 1][ml];
      acc = wmma_f32x4(a, b, acc);
    }
    wait_async0();        // next chunk's async copies have landed
    __syncthreads();      // all waves done reading `cur`, data visible
  }

  const int h  = lane & 15;
  const int mh = (lane >> 4) << 3;   // 0 or 8
  const float bb = bred[h];
#pragma unroll
  for (int r = 0; r < 8; ++r) {
    int grow = blockM + wave * 16 + mh + r;   // global row = b*L + l
    int bidx = grow >> 12;                    // / 4096
    int l    = grow & (LL - 1);
    out[((size_t)(bidx * HH + h)) * LL + l] = acc[r] + bb;
  }
}

// ---------------------------------------------------------------------------
// 3) Full GEMM with WMMA + async double-buffered LDS staging
//    Y = X @ W + bias    (MM x DD) @ (DD x DD)
//    block tile 128(M) x 128(N), KC2=16; 8 waves as 4x2, each wave 2x4 tiles
//    grid (DD/128, MM/128) x 256 threads
// ---------------------------------------------------------------------------
#define KC2 16
#define PX  20    // sX row stride (80B: 16B aligned, conflict-free columns)
#define PW  136   // sW row stride (544B: 16B aligned, half-waves bank-disjoint)

__global__ void gemm_bias_wmma(const float* __restrict__ X,
                               const float* __restrict__ W,
                               const float* __restrict__ bias,
                               float* __restrict__ Y) {
  __shared__ float sX[2][128][PX];   // 20.5 KB
  __shared__ float sW[2][KC2][PW];   // 17.4 KB

  const int tid  = threadIdx.x;
  const int lane = tid & 31;
  const int wave = tid >> 5;
  const int wm   = wave >> 1;      // 0..3  -> 32 rows
  const int wn   = wave & 1;       // 0..1  -> 64 cols
  const int blockM = blockIdx.y * 128;
  const int blockN = blockIdx.x * 128;

  const int ml   = lane & 15;
  const int koff = (lane >> 4) << 1;

  // per-thread staging assignments (4 async B128 per chunk, all lanes issue)
  const int xr0 = tid >> 2,         xc0 = (tid & 3) * 4;       // X: rows 0..63
  const int xr1 = (tid + 256) >> 2, xc1 = (tid & 3) * 4;       // X: rows 64..127
  const int wr0 = tid >> 5,         wc0 = (tid & 31) * 4;      // W: rows 0..7
  const int wr1 = (tid + 256) >> 5, wc1 = (tid & 31) * 4;      // W: rows 8..15

  v8f acc[2][4] = {{{}, {}, {}, {}}, {{}, {}, {}, {}}};

  const int nchunks = DD / KC2;

  // prologue: stage chunk 0 into buffer 0
  {
    const int k0 = 0;
    async_load_b128((unsigned)(uintptr_t)&sX[0][xr0][xc0],
                    &X[(size_t)(blockM + xr0) * DD + k0 + xc0]);
    async_load_b128((unsigned)(uintptr_t)&sX[0][xr1][xc1],
                    &X[(size_t)(blockM + xr1) * DD + k0 + xc1]);
    async_load_b128((unsigned)(uintptr_t)&sW[0][wr0][wc0],
                    &W[(size_t)(k0 + wr0) * DD + blockN + wc0]);
    async_load_b128((unsigned)(uintptr_t)&sW[0][wr1][wc1],
                    &W[(size_t)(k0 + wr1) * DD + blockN + wc1]);
  }
  wait_async0();
  __syncthreads();

  for (int c = 0; c < nchunks; ++c) {
    const int cur = c & 1;
    if (c + 1 < nchunks) {
      const int nb = cur ^ 1;
      const int k0 = (c + 1) * KC2;
      async_load_b128((unsigned)(uintptr_t)&sX[nb][xr0][xc0],
                      &X[(size_t)(blockM + xr0) * DD + k0 + xc0]);
      async_load_b128((unsigned)(uintptr_t)&sX[nb][xr1][xc1],
                      &X[(size_t)(blockM + xr1) * DD + k0 + xc1]);
      async_load_b128((unsigned)(uintptr_t)&sW[nb][wr0][wc0],
                      &W[(size_t)(k0 + wr0) * DD + blockN + wc0]);
      async_load_b128((unsigned)(uintptr_t)&sW[nb][wr1][wc1],
                      &W[(size_t)(k0 + wr1) * DD + blockN + wc1]);
    }
#pragma unroll
    for (int ks = 0; ks < KC2; ks += 4) {
      v2f a[2], b[4];
#pragma unroll
      for (int mi = 0; mi < 2; ++mi) {
        int row = wm * 32 + mi * 16 + ml;
        a[mi].x = sX[cur][row][ks + koff];
        a[mi].y = sX[cur][row][ks + koff + 1];
      }
#pragma unroll
      for (int ni = 0; ni < 4; ++ni) {
        int col = wn * 64 + ni * 16 + ml;
        b[ni].x = sW[cur][ks + koff][col];
        b[ni].y = sW[cur][ks + koff + 1][col];
      }
#pragma unroll
      for (int mi = 0; mi < 2; ++mi)
#pragma unroll
        for (int ni = 0; ni < 4; ++ni)
          acc[mi][ni] = wmma_f32x4(a[mi], b[ni], acc[mi][ni]);
    }
    wait_async0();        // next chunk's async copies have landed
    __syncthreads();      // all waves done reading `cur`, data visible
  }

  const int nl = lane & 15;
  const int mh = (lane >> 4) << 3;
#pragma unroll
  for (int mi = 0; mi < 2; ++mi) {
#pragma unroll
    for (int ni = 0; ni < 4; ++ni) {
      int col = blockN + wn * 64 + ni * 16 + nl;
      float bv = bias[col];
#pragma unroll
      for (int r = 0; r < 8; ++r) {
        int row = blockM + wm * 32 + mi * 16 + mh + r;
        Y[(size_t)row * DD + col] = acc[mi][ni][r] + bv;
      }
    }
  }
}

// ---------------------------------------------------------------------------
// 4) Circular autocorrelation, direct from LDS:
//    corr[bh][d] = sum_t qm[bh][(t+d)&4095] * km[bh][t]
// ---------------------------------------------------------------------------
__global__ void corr_kernel(const float* __restrict__ qm,
                            const float* __restrict__ km,
                            float* __restrict__ corr) {
  __shared__ float sq[LL];
  __shared__ float sk[LL];
  const int bh  = blockIdx.x;
  const int tid = threadIdx.x;
  for (int i = tid; i < LL; i += 256) {
    sq[i] = qm[(size_t)bh * LL + i];
    sk[i] = km[(size_t)bh * LL + i];
  }
  __syncthreads();
  const int d = blockIdx.y * 256 + tid;
  float s = 0.f;
#pragma unroll 4
  for (int t = 0; t < LL; ++t) s += sq[(t + d) & (LL - 1)] * sk[t];
  corr[(size_t)bh * LL + d] = s;
}

// ---------------------------------------------------------------------------
// 5) Top-8 lags + softmax weights per (b,h). grid 64 x 256 threads.
// ---------------------------------------------------------------------------
__global__ void topk_softmax_kernel(const float* __restrict__ corr,
                                    float* __restrict__ wts,
                                    int* __restrict__ delays) {
  __shared__ float sc[LL];
  __shared__ float rv[256];
  __shared__ int   ri[256];
  __shared__ float topv[TOPK];
  __shared__ int   topi[TOPK];
  const int bh  = blockIdx.x;
  const int tid = threadIdx.x;
  for (int i = tid; i < LL; i += 256) sc[i] = corr[(size_t)bh * LL + i];
  __syncthreads();

  for (int k = 0; k < TOPK; ++k) {
    float best = -INFINITY; int bi = 0;
    for (int i = tid; i < LL; i += 256) {
      float x = sc[i];
      if (x > best) { best = x; bi = i; }
    }
    rv[tid] = best; ri[tid] = bi;
    __syncthreads();
    for (int s = 128; s > 0; s >>= 1) {
      if (tid < s && rv[tid + s] > rv[tid]) { rv[tid] = rv[tid + s]; ri[tid] = ri[tid + s]; }
      __syncthreads();
    }
    if (tid == 0) { topv[k] = rv[0]; topi[k] = ri[0]; sc[ri[0]] = -INFINITY; }
    __syncthreads();
  }

  if (tid == 0) {
    float mx = topv[0];       // sorted descending
    float e[TOPK], s = 0.f;
#pragma unroll
    for (int k = 0; k < TOPK; ++k) { e[k] = __expf(topv[k] - mx); s += e[k]; }
    float invs = 1.f / s;
#pragma unroll
    for (int k = 0; k < TOPK; ++k) {
      wts[bh * TOPK + k]    = e[k] * invs;
      delays[bh * TOPK + k] = topi[k];
    }
  }
}

// ---------------------------------------------------------------------------
// 6) Weighted circular rolls of V -> attn (B,L,D) merged-head layout
// ---------------------------------------------------------------------------
__global__ void roll_accum_kernel(const float* __restrict__ v,
                                  const float* __restrict__ wts,
                                  const int* __restrict__ delays,
                                  float* __restrict__ attn) {
  const size_t idx = (size_t)blockIdx.x * 256 + threadIdx.x;  // < B*L*D
  const int c  = (int)(idx & (DD - 1));
  const int h  = c >> 6;
  const int gl = (int)(idx >> 10);       // b*L + l
  const int b  = gl >> 12;
  const int l  = gl & (LL - 1);
  const int base = (b * HH + h) * TOPK;
  float s = 0.f;
#pragma unroll
  for (int k = 0; k < TOPK; ++k) {
    const float w = wts[base + k];
    const int   d = delays[base + k];
    const int lsrc = (l - d) & (LL - 1);
    s += w * v[((size_t)(b * LL + lsrc)) * DD + c];
  }
  attn[idx] = s;
}

// ---------------------------------------------------------------------------
// launch
// ---------------------------------------------------------------------------
extern "C" void kernel_launch(void* const* d_in, const int* in_sizes, int n_in,
                              void* d_out, int out_size, void* d_ws, size_t ws_size,
                              hipStream_t stream) {
  const float* queries = (const float*)d_in[0];
  const float* keys    = (const float*)d_in[1];
  const float* values  = (const float*)d_in[2];
  const float* Wq = (const float*)d_in[3];  const float* bq = (const float*)d_in[4];
  const float* Wk = (const float*)d_in[5];  const float* bk = (const float*)d_in[6];
  const float* Wv = (const float*)d_in[7];  const float* bv = (const float*)d_in[8];
  const float* Wo = (const float*)d_in[9];  const float* bo = (const float*)d_in[10];
  float* out = (float*)d_out;

  // workspace carve-up (~137 MB)
  float* ws   = (float*)d_ws;
  float* vbuf = ws;                              // B*L*D = 16777216
  float* attn = vbuf + (size_t)MM * DD;          // 16777216
  float* qm   = attn + (size_t)MM * DD;          // B*H*L = 262144
  float* km   = qm + (size_t)BB * HH * LL;
  float* corr = km + (size_t)BB * HH * LL;
  float* Wqr  = corr + (size_t)BB * HH * LL;     // 1024*16
  float* Wkr  = Wqr + DD * HH;
  float* bqr  = Wkr + DD * HH;                   // 16
  float* bkr  = bqr + HH;
  float* wts  = bkr + HH;                        // B*H*8 = 512
  int*   dly  = (int*)(wts + BB * HH * TOPK);    // 512 ints

  // 1) head-mean-collapsed projection weights for q and k
  reduce_w_kernel<<<64, 256, 0, stream>>>(Wq, bq, Wqr, bqr);
  reduce_w_kernel<<<64, 256, 0, stream>>>(Wk, bk, Wkr, bkr);

  // 2) qm / km via narrow WMMA GEMM (N = 16 heads, async staging)
  gemm_mean_wmma<<<MM / 128, 256, 0, stream>>>(queries, Wqr, bqr, qm);
  gemm_mean_wmma<<<MM / 128, 256, 0, stream>>>(keys,    Wkr, bkr, km);

  // 3) full V projection (WMMA fp32, async double-buffered staging)
  gemm_bias_wmma<<<dim3(DD / 128, MM / 128), 256, 0, stream>>>(values, Wv, bv, vbuf);

  // 4) autocorrelation, 5) top-k + softmax
  corr_kernel<<<dim3(BB * HH, LL / 256), 256, 0, stream>>>(qm, km, corr);
  topk_softmax_kernel<<<BB * HH, 256, 0, stream>>>(corr, wts, dly);

  // 6) weighted circular rolls
  roll_accum_kernel<<<(MM * DD) / 256, 256, 0, stream>>>(vbuf, wts, dly, attn);

  // 7) output projection (WMMA fp32, async double-buffered staging)
  gemm_bias_wmma<<<dim3(DD / 128, MM / 128), 256, 0, stream>>>(attn, Wo, bo, out);
}